// fk_layer_57174604644767
// MI455X (gfx1250) — compile-verified
//
#include <hip/hip_runtime.h>
#include <stdint.h>

#define NJ   31
#define EPB  64            // batch elements per block
#define TPB  64            // threads per block (2 independent waves, wave32)
#define WPE  32            // elements per wave
#define FPE  124           // floats per element for joint & output (31*4)

// ---------- CDNA5 async global<->LDS helpers (ASYNCcnt-tracked, GVS addressing) ----------
// LDS side addressed by VDST, global side by SADDR + VADDR; same imm offset added to both sides

__device__ __forceinline__ unsigned lds_off_u32(const void* p) {
  // flat LDS address keeps the workgroup-relative byte offset in bits [31:0]
  return (unsigned)(uintptr_t)p;
}

template<int I, int N, int STEP, int OFS>
struct Rep {
  static __device__ __forceinline__ void ld128(unsigned lds, unsigned voff, unsigned long long base) {
    asm volatile("global_load_async_to_lds_b128 %0, %1, %2 offset:%3"
                 :: "v"(lds), "v"(voff), "s"(base), "n"(OFS + I * STEP) : "memory");
    Rep<I + 1, N, STEP, OFS>::ld128(lds, voff, base);
  }
  static __device__ __forceinline__ void ld32(unsigned lds, unsigned voff, unsigned long long base) {
    asm volatile("global_load_async_to_lds_b32 %0, %1, %2 offset:%3"
                 :: "v"(lds), "v"(voff), "s"(base), "n"(OFS + I * STEP) : "memory");
    Rep<I + 1, N, STEP, OFS>::ld32(lds, voff, base);
  }
  static __device__ __forceinline__ void st128(unsigned voff, unsigned lds, unsigned long long base) {
    asm volatile("global_store_async_from_lds_b128 %0, %1, %2 offset:%3"
                 :: "v"(voff), "v"(lds), "s"(base), "n"(OFS + I * STEP) : "memory");
    Rep<I + 1, N, STEP, OFS>::st128(voff, lds, base);
  }
};
template<int N, int STEP, int OFS>
struct Rep<N, N, STEP, OFS> {
  static __device__ __forceinline__ void ld128(unsigned, unsigned, unsigned long long) {}
  static __device__ __forceinline__ void ld32 (unsigned, unsigned, unsigned long long) {}
  static __device__ __forceinline__ void st128(unsigned, unsigned, unsigned long long) {}
};

__device__ __forceinline__ void wait_async0() {
  asm volatile("s_wait_asynccnt 0x0" ::: "memory");
}
__device__ __forceinline__ void wait_ds0() {
  asm volatile("s_wait_dscnt 0x0" ::: "memory");
}

// ---------- small affine algebra (rotation 3x3 + translation, bottom row [0001]) ----------

struct Aff { float r[9]; float t[3]; };

__device__ __forceinline__ Aff compose(const Aff& A, const Aff& B) {
  Aff C;
#pragma unroll
  for (int i = 0; i < 3; ++i) {
    const float a0 = A.r[3*i+0], a1 = A.r[3*i+1], a2 = A.r[3*i+2];
    C.r[3*i+0] = a0*B.r[0] + a1*B.r[3] + a2*B.r[6];
    C.r[3*i+1] = a0*B.r[1] + a1*B.r[4] + a2*B.r[7];
    C.r[3*i+2] = a0*B.r[2] + a1*B.r[5] + a2*B.r[8];
    C.t[i]     = a0*B.t[0] + a1*B.t[1] + a2*B.t[2] + A.t[i];
  }
  return C;
}

// Build RT[j] = R4(quat_j) @ T(offsets_j, bone_j). `offs` is grid-uniform -> scalar loads.
__device__ __forceinline__ Aff make_rt(int j, const float4* quat4,
                                       const float* boneRow,
                                       const float* __restrict__ offs) {
  const float4 q = quat4[j];
  const float w = q.x, x = q.y, y = q.z, z = q.w;
  const float s = 2.0f / (w*w + x*x + y*y + z*z);
  float R[9];
  R[0] = 1.0f - s*(y*y + z*z); R[1] = s*(x*y - z*w);        R[2] = s*(x*z + y*w);
  R[3] = s*(x*y + z*w);        R[4] = 1.0f - s*(x*x + z*z); R[5] = s*(y*z - x*w);
  R[6] = s*(x*z - y*w);        R[7] = s*(y*z + x*w);        R[8] = 1.0f - s*(x*x + y*y);

  const float bl = boneRow[j];
  const float* O = offs + j*16;          // row-major 4x4, bottom row [0,0,0,1]
  const float t0 = O[3]*bl, t1 = O[7]*bl, t2 = O[11]*bl;   // scale first, like reference

  Aff RT;
#pragma unroll
  for (int i = 0; i < 3; ++i) {
    const float a0 = R[3*i+0], a1 = R[3*i+1], a2 = R[3*i+2];
    RT.r[3*i+0] = a0*O[0] + a1*O[4] + a2*O[8];
    RT.r[3*i+1] = a0*O[1] + a1*O[5] + a2*O[9];
    RT.r[3*i+2] = a0*O[2] + a1*O[6] + a2*O[10];
    RT.t[i]     = a0*t0   + a1*t1   + a2*t2;   // R4 translation is 0
  }
  return RT;
}

// ---------- kernel ----------

__global__ __launch_bounds__(TPB)
void fk_kernel(const float* __restrict__ rootp,   // (B,3)
               const float* __restrict__ joint,   // (B,31*4)
               const float* __restrict__ bone,    // (B,31)
               const float* __restrict__ offs,    // (31,4,4)
               float* __restrict__ out)           // (B,31,4)
{
  __shared__ __align__(16) float jointS[EPB * FPE]; // quats in, results out (in place)
  __shared__ __align__(16) float boneS [EPB * NJ];
  __shared__ __align__(16) float rootS [EPB * 3];

  const int tid  = threadIdx.x;
  const int lane = tid & 31;
  const long long blk = blockIdx.x;

  // wave-uniform byte offsets of this wave's slice, forced onto the scalar path
  const unsigned wJ = (unsigned)__builtin_amdgcn_readfirstlane((tid >> 5) * (WPE * FPE * 4)); // 15872
  const unsigned wB = (unsigned)__builtin_amdgcn_readfirstlane((tid >> 5) * (WPE * NJ  * 4)); // 3968
  const unsigned wR = (unsigned)__builtin_amdgcn_readfirstlane((tid >> 5) * (WPE * 3   * 4)); // 384

  const unsigned long long jointW = (unsigned long long)(uintptr_t)(joint + blk * (EPB * FPE)) + wJ;
  const unsigned long long boneW  = (unsigned long long)(uintptr_t)(bone  + blk * (EPB * NJ))  + wB;
  const unsigned long long rootW  = (unsigned long long)(uintptr_t)(rootp + blk * (EPB * 3))   + wR;
  const unsigned long long outW   = (unsigned long long)(uintptr_t)(out   + blk * (EPB * FPE)) + wJ;

  const unsigned v16 = (unsigned)lane * 16u;  // per-lane byte offset for b128 copies
  const unsigned v4  = (unsigned)lane * 4u;   // per-lane byte offset for b32 copies
  const unsigned jL  = lds_off_u32(jointS) + wJ;
  const unsigned bL  = lds_off_u32(boneS)  + wB;
  const unsigned rL  = lds_off_u32(rootS)  + wR;

  // --- per-wave coalesced async staging of this wave's 32 elements (44 ops < ASYNCcnt max) ---
  Rep<0, 31, 32 * 16, 0   >::ld128(jL + v16, v16, jointW); // 32*496 B linear
  Rep<0,  7, 32 * 16, 0   >::ld128(bL + v16, v16, boneW);  // first 3584 B of 3968
  Rep<0,  3, 32 *  4, 3584>::ld32 (bL + v4,  v4,  boneW);  // 384 B tail
  Rep<0,  3, 32 *  4, 0   >::ld32 (rL + v4,  v4,  rootW);  // 32*12 B
  wait_async0();                     // this wave's LDS slice is ready; no barrier needed

  // --- per-thread FK over the fixed tree ---
  float4*      quat4   = (float4*)(jointS + tid * FPE);
  const float* boneRow = boneS + tid * NJ;
  const float  rx = rootS[tid*3+0] * 100.0f;
  const float  ry = rootS[tid*3+1] * 100.0f;
  const float  rz = rootS[tid*3+2] * 100.0f;

  Aff run, m0, m13;

  // j = 0 : mats[0] = root_T @ RT0  (root rotation = I)
  run = make_rt(0, quat4, boneRow, offs);
  run.t[0] += rx; run.t[1] += ry; run.t[2] += rz;
  quat4[0] = make_float4(run.t[0], run.t[1], run.t[2], 1.0f);
  m0 = run;

#pragma unroll
  for (int j = 1; j <= 5; ++j) {                 // chain 1..5
    run = compose(run, make_rt(j, quat4, boneRow, offs));
    quat4[j] = make_float4(run.t[0], run.t[1], run.t[2], 1.0f);
  }
  run = m0;
#pragma unroll
  for (int j = 6; j <= 10; ++j) {                // chain 6..10 (parent 0)
    run = compose(run, make_rt(j, quat4, boneRow, offs));
    quat4[j] = make_float4(run.t[0], run.t[1], run.t[2], 1.0f);
  }
  run = m0;
#pragma unroll
  for (int j = 11; j <= 15; ++j) {               // chain 11..15 (parent 0); save mats[13]
    run = compose(run, make_rt(j, quat4, boneRow, offs));
    quat4[j] = make_float4(run.t[0], run.t[1], run.t[2], 1.0f);
    if (j == 13) m13 = run;
  }
  run = m13;
#pragma unroll
  for (int j = 16; j <= 20; ++j) {               // chains rooted at mats[13]
    run = compose(run, make_rt(j, quat4, boneRow, offs));
    quat4[j] = make_float4(run.t[0], run.t[1], run.t[2], 1.0f);
  }
  run = m13;
#pragma unroll
  for (int j = 21; j <= 25; ++j) {
    run = compose(run, make_rt(j, quat4, boneRow, offs));
    quat4[j] = make_float4(run.t[0], run.t[1], run.t[2], 1.0f);
  }
  run = m13;
#pragma unroll
  for (int j = 26; j <= 30; ++j) {
    run = compose(run, make_rt(j, quat4, boneRow, offs));
    quat4[j] = make_float4(run.t[0], run.t[1], run.t[2], 1.0f);
  }

  // --- per-wave coalesced async drain: LDS -> global output ---
  wait_ds0();                        // this wave's result ds_stores are in LDS
  Rep<0, 31, 32 * 16, 0>::st128(v16, jL + v16, outW);
  wait_async0();
}

extern "C" void kernel_launch(void* const* d_in, const int* in_sizes, int n_in,
                              void* d_out, int out_size, void* d_ws, size_t ws_size,
                              hipStream_t stream) {
  const float* rootp = (const float*)d_in[0];   // (B,3)
  const float* joint = (const float*)d_in[1];   // (B,124)
  const float* bone  = (const float*)d_in[2];   // (B,31)
  const float* offs  = (const float*)d_in[3];   // (31,4,4)
  float* out = (float*)d_out;                   // (B,31,4)

  const int batch  = in_sizes[0] / 3;           // 131072
  const int blocks = batch / EPB;               // 2048

  fk_kernel<<<dim3(blocks), dim3(TPB), 0, stream>>>(rootp, joint, bone, offs, out);
}